// ExpanderSimpleGraphSageLayer_44470091382964
// MI455X (gfx1250) — compile-verified
//
#include <hip/hip_runtime.h>
#include <math.h>

// Problem constants (match the reference).
#define NN 100000
#define DD 128
#define L2EPS 1e-12f

typedef __attribute__((ext_vector_type(2))) float v2f;
typedef __attribute__((ext_vector_type(8))) float v8f;

// ---------------------------------------------------------------------------
// K0: zero the scatter accumulator (N*D floats) + degree array (N floats).
// Workspace is poisoned / left dirty between calls, so this runs every call.
// ---------------------------------------------------------------------------
__global__ __launch_bounds__(256) void k_zero(float4* __restrict__ p, int n4) {
  int i = blockIdx.x * blockDim.x + threadIdx.x;
  if (i < n4) p[i] = make_float4(0.f, 0.f, 0.f, 0.f);
}

// ---------------------------------------------------------------------------
// K1: edge scatter. One wave32 per edge; lane L owns float4 [4L..4L+3] of the
// 128-wide feature row.  msg = h[src] * norm[src], atomically accumulated
// into cacc[dst]; lane 0 counts the in-degree.  h (51MB) and cacc (51MB) are
// L2-resident on MI455X (192MB L2), so gathers/atomics stay off HBM.
// ---------------------------------------------------------------------------
__global__ __launch_bounds__(256) void k_edge(const float* __restrict__ h,
                                              const float* __restrict__ norm,
                                              const int* __restrict__ esrc,
                                              const int* __restrict__ edst,
                                              float* __restrict__ cacc,
                                              float* __restrict__ deg,
                                              int E) {
  unsigned tid = blockIdx.x * blockDim.x + threadIdx.x;
  unsigned wave = tid >> 5;
  unsigned lane = threadIdx.x & 31u;
  if (wave >= (unsigned)E) return;   // wave-uniform guard
  int s = esrc[wave];
  int d = edst[wave];
  float ns = norm[s];
  const float4* hrow = reinterpret_cast<const float4*>(h + (size_t)s * DD);
  float4 v = hrow[lane];
  float* crow = cacc + (size_t)d * DD + (size_t)lane * 4u;
  atomicAdd(crow + 0, v.x * ns);
  atomicAdd(crow + 1, v.y * ns);
  atomicAdd(crow + 2, v.z * ns);
  atomicAdd(crow + 3, v.w * ns);
  if (lane == 0u) atomicAdd(deg + d, 1.0f);
}

// ---------------------------------------------------------------------------
// K2a: finalize mean c = cacc / max(deg,1); write h_out = h + c*norm, and
// stage c into the second half of each b_out row (cols 128..255) so K2b can
// consume it without extra workspace.
// ---------------------------------------------------------------------------
__global__ __launch_bounds__(256) void k_node_a(const float* __restrict__ h,
                                                const float* __restrict__ norm,
                                                const float* __restrict__ cacc,
                                                const float* __restrict__ deg,
                                                float* __restrict__ hout,
                                                float* __restrict__ bout) {
  unsigned i = blockIdx.x * blockDim.x + threadIdx.x;
  if (i >= (unsigned)(NN * (DD / 4))) return;
  size_t e = (size_t)i * 4u;
  unsigned node = (unsigned)(e >> 7);   // e / 128
  unsigned col  = (unsigned)(e & 127u);
  float inv = 1.0f / fmaxf(deg[node], 1.0f);
  float4 c = *reinterpret_cast<const float4*>(cacc + e);
  c.x *= inv; c.y *= inv; c.z *= inv; c.w *= inv;
  float nm = norm[node];
  float4 hv = *reinterpret_cast<const float4*>(h + e);
  float4 ho;
  ho.x = hv.x + c.x * nm;
  ho.y = hv.y + c.y * nm;
  ho.z = hv.z + c.z * nm;
  ho.w = hv.w + c.w * nm;
  *reinterpret_cast<float4*>(hout + e) = ho;
  *reinterpret_cast<float4*>(bout + (size_t)node * (2 * DD) + DD + col) = c;
}

// ---------------------------------------------------------------------------
// K2b: L2-normalize bundle = concat(b, c) per node.
// One wave32 per 16-node tile. Row sums of squares are computed with a chain
// of V_WMMA_F32_16X16X4_F32 against an all-ones B matrix:
//   D = A(16x4) * ones(4x16) + C  ->  every column of D holds the row sum.
// A layout (32-bit A, 16x4): lane L (row L&15) holds K = {0,1} if L<16 else
// {2,3}  ->  load float2 at column 4k + (L<16 ? 0 : 2).
// D layout: VGPR v at lane L = row (v + 8*(L>=16)); identical across columns,
// so each lane extracts its row's sum with an 8-way select.
// ---------------------------------------------------------------------------
__global__ __launch_bounds__(256) void k_node_b(const float* __restrict__ b,
                                                float* __restrict__ bout,
                                                int ntiles) {
  unsigned tid = blockIdx.x * blockDim.x + threadIdx.x;
  unsigned wave = tid >> 5;
  unsigned lane = threadIdx.x & 31u;
  if (wave >= (unsigned)ntiles) return;  // wave-uniform guard; EXEC stays full
  const unsigned row0 = wave * 16u;
  const unsigned r    = lane & 15u;
  const unsigned koff = (lane < 16u) ? 0u : 2u;

  const float* brow = b    + (size_t)(row0 + r) * DD + koff;
  const float* crow = bout + (size_t)(row0 + r) * (2 * DD) + DD + koff; // staged c

  v8f acc = {};
  v2f bones;
  bones[0] = 1.0f; bones[1] = 1.0f;

  // First 128 bundle columns: b
#pragma unroll
  for (int k = 0; k < 32; ++k) {
    v2f x = *reinterpret_cast<const v2f*>(brow + 4 * k);
    v2f a = x * x;
    acc = __builtin_amdgcn_wmma_f32_16x16x4_f32(
        /*neg_a=*/false, a, /*neg_b=*/false, bones,
        /*c_mod=*/(short)0, acc, /*reuse_a=*/false, /*reuse_b=*/false);
  }
  // Last 128 bundle columns: staged c
#pragma unroll
  for (int k = 0; k < 32; ++k) {
    v2f x = *reinterpret_cast<const v2f*>(crow + 4 * k);
    v2f a = x * x;
    acc = __builtin_amdgcn_wmma_f32_16x16x4_f32(
        false, a, false, bones, (short)0, acc, false, false);
  }

  // Lane L owns output row rw = 8*(L>=16) + (L&7); its sum is acc[L&7].
  unsigned k7 = lane & 7u;
  float ss = acc[0];
  ss = (k7 == 1u) ? acc[1] : ss;
  ss = (k7 == 2u) ? acc[2] : ss;
  ss = (k7 == 3u) ? acc[3] : ss;
  ss = (k7 == 4u) ? acc[4] : ss;
  ss = (k7 == 5u) ? acc[5] : ss;
  ss = (k7 == 6u) ? acc[6] : ss;
  ss = (k7 == 7u) ? acc[7] : ss;
  float inv = 1.0f / fmaxf(sqrtf(ss), L2EPS);

  unsigned rw   = ((lane >> 4) & 1u) * 8u + k7;   // row within tile
  unsigned half = (lane >> 3) & 1u;               // which 128-col half
  size_t orow = (size_t)(row0 + rw) * (2 * DD);

  if (half == 0u) {
    // cols 0..127: source is b, scaled into b_out
    const float4* sp = reinterpret_cast<const float4*>(b + (size_t)(row0 + rw) * DD);
    float4* dp = reinterpret_cast<float4*>(bout + orow);
#pragma unroll
    for (int j = 0; j < 32; ++j) {
      float4 v = sp[j];
      v.x *= inv; v.y *= inv; v.z *= inv; v.w *= inv;
      dp[j] = v;
    }
  } else {
    // cols 128..255: staged c, scaled in place
    float* sp = bout + orow + DD;
#pragma unroll
    for (int j = 0; j < 32; ++j) {
      float4 v = *reinterpret_cast<float4*>(sp + 4 * j);
      v.x *= inv; v.y *= inv; v.z *= inv; v.w *= inv;
      *reinterpret_cast<float4*>(sp + 4 * j) = v;
    }
  }
}

// ---------------------------------------------------------------------------
// Launch: inputs are (h, b, norm, edge_src, edge_dst); output is
// [h_out (N*128) | b_out (N*256)] flat.  Workspace: cacc (N*128 f32) + deg
// (N f32) = ~51.6 MB.
// ---------------------------------------------------------------------------
extern "C" void kernel_launch(void* const* d_in, const int* in_sizes, int n_in,
                              void* d_out, int out_size, void* d_ws, size_t ws_size,
                              hipStream_t stream) {
  const float* h    = (const float*)d_in[0];
  const float* b    = (const float*)d_in[1];
  const float* norm = (const float*)d_in[2];
  const int*   esrc = (const int*)d_in[3];
  const int*   edst = (const int*)d_in[4];
  const int E = in_sizes[3];

  float* hout = (float*)d_out;
  float* bout = hout + (size_t)NN * DD;

  float* cacc = (float*)d_ws;
  float* deg  = cacc + (size_t)NN * DD;

  // K0: zero cacc + deg (contiguous, (N*D + N) floats, divisible by 4)
  const int n4 = (NN * DD + NN) / 4;
  k_zero<<<(n4 + 255) / 256, 256, 0, stream>>>((float4*)cacc, n4);

  // K1: one wave per edge
  {
    long long threads = (long long)E * 32;
    int blocks = (int)((threads + 255) / 256);
    k_edge<<<blocks, 256, 0, stream>>>(h, norm, esrc, edst, cacc, deg, E);
  }

  // K2a: finalize c, write h_out, stage c into b_out[:,128:256]
  {
    int threads = NN * (DD / 4);
    k_node_a<<<(threads + 255) / 256, 256, 0, stream>>>(h, norm, cacc, deg, hout, bout);
  }

  // K2b: WMMA row-sum-of-squares + scale, one wave per 16-node tile
  {
    int ntiles = NN / 16;  // 6250, exact
    long long threads = (long long)ntiles * 32;
    int blocks = (int)((threads + 255) / 256);
    k_node_b<<<blocks, 256, 0, stream>>>(b, bout, ntiles);
  }
}